// NeuralNetwork_9569187136204
// MI455X (gfx1250) — compile-verified
//
#include <hip/hip_runtime.h>

#define NBLK 128
#define BLKT 256
#define NN   1024
#define RS   1025               // weight row stride (bias in last column)
#define NLAYER 64
#define TSTR 68                 // LDS tile row stride (floats): bank-conflict-free

typedef __attribute__((ext_vector_type(8))) float v8f;
typedef __attribute__((ext_vector_type(2))) float v2f;

__device__ __forceinline__ float wave_sum(float v) {
#pragma unroll
  for (int o = 16; o > 0; o >>= 1) v += __shfl_down(v, o);
  return v;
}
__device__ __forceinline__ float wave_max(float v) {
#pragma unroll
  for (int o = 16; o > 0; o >>= 1) v = fmaxf(v, __shfl_down(v, o));
  return v;
}
__device__ __forceinline__ float wave_min(float v) {
#pragma unroll
  for (int o = 16; o > 0; o >>= 1) v = fminf(v, __shfl_down(v, o));
  return v;
}

// Monotonic-ticket grid barrier. Counter is zeroed by hipMemsetAsync each call.
__device__ __forceinline__ void grid_barrier(unsigned int* bar) {
  __syncthreads();
  if (threadIdx.x == 0) {
    __threadfence();
    unsigned int t = __hip_atomic_fetch_add(bar, 1u, __ATOMIC_ACQ_REL,
                                            __HIP_MEMORY_SCOPE_AGENT);
    unsigned int target = (t / NBLK + 1u) * (unsigned int)NBLK;
    while (__hip_atomic_load(bar, __ATOMIC_ACQUIRE,
                             __HIP_MEMORY_SCOPE_AGENT) < target) {
      __builtin_amdgcn_s_sleep(2);
    }
  }
  __syncthreads();
}

// 16-row x 64-k fp32 GEMV partial using V_WMMA_F32_16X16X4_F32, with the weight
// tile staged Global->LDS by coalesced global_load_async_to_lds_b32 (128B
// contiguous per instruction, ASYNCcnt-tracked), issued column-group-major so
// WMMA on columns [0,32) overlaps the fill of columns [32,64).
// A tile in LDS (stride TSTR=68 floats): lane m=lane&15 reads row m at banks
// 4m/4m+1 (lanes 0-15, k=kh) and 4m+2/4m+3 (lanes 16-31, k=kh+2..3) -> no
// bank conflicts. B = vector chunk broadcast into all 16 columns, so every D
// column holds the 16 dot products; lanes 0/16 own rows row0..+7 / +8..+15.
__device__ __forceinline__ void gemv16(const float* __restrict__ W, int row0,
                                       int kb, float* __restrict__ tile,
                                       const float* __restrict__ vloc,
                                       float* __restrict__ out, bool addBias) {
  const int lane = threadIdx.x & 31;
  const unsigned tlds = (unsigned)(uintptr_t)tile;  // LDS byte offset (low 32 bits)
#pragma unroll
  for (int s = 0; s < 2; ++s) {
#pragma unroll
    for (int r = 0; r < 16; ++r) {
      const float* gp = W + (size_t)(row0 + r) * RS + (size_t)(kb + s * 32 + lane);
      unsigned la = tlds + (unsigned)((r * TSTR + s * 32 + lane) * 4);
      asm volatile("global_load_async_to_lds_b32 %0, %1, off"
                   :: "v"(la), "v"(gp) : "memory");
    }
  }
  const int m  = lane & 15;
  const int kh = (lane >> 4) << 1;                  // 0 or 2
  const float* arow = tile + m * TSTR + kh;
  const float* bcol = vloc + kb + kh;
  v8f c = {0.f, 0.f, 0.f, 0.f, 0.f, 0.f, 0.f, 0.f};

  asm volatile("s_wait_asynccnt 16" ::: "memory");  // first 16 asyncs (cols 0..31) done
#pragma unroll
  for (int kk = 0; kk < 32; kk += 4) {
    v2f a = *(const v2f*)(arow + kk);
    v2f b = *(const v2f*)(bcol + kk);
    c = __builtin_amdgcn_wmma_f32_16x16x4_f32(false, a, false, b,
                                              (short)0, c, false, false);
  }
  asm volatile("s_wait_asynccnt 0" ::: "memory");   // cols 32..63 resident
#pragma unroll
  for (int kk = 32; kk < 64; kk += 4) {
    v2f a = *(const v2f*)(arow + kk);
    v2f b = *(const v2f*)(bcol + kk);
    c = __builtin_amdgcn_wmma_f32_16x16x4_f32(false, a, false, b,
                                              (short)0, c, false, false);
  }
  asm volatile("s_wait_dscnt 0x0" ::: "memory");    // tile reusable by next task

  if ((lane & 15) == 0) {                           // lanes 0 and 16
    int rbase = row0 + ((lane >> 4) << 3);
#pragma unroll
    for (int g = 0; g < 8; ++g) {
      float val = c[g];
      if (addBias) val += W[(size_t)(rbase + g) * RS + (RS - 1)];
      atomicAdd(&out[rbase + g], val);
    }
  }
}

__global__ void __launch_bounds__(BLKT, 1)
net_kernel(const float* __restrict__ w_b, const float* __restrict__ attn,
           const float* __restrict__ nrm, const float* __restrict__ ada,
           float* __restrict__ dout, float* __restrict__ ws,
           unsigned int* __restrict__ bar) {
  // workspace layout (floats): v | q | k | u | v2 | z   (q,k,u contiguous)
  float* v   = ws;
  float* q   = ws + 1 * NN;
  float* kk_ = ws + 2 * NN;
  float* u   = ws + 3 * NN;
  float* v2  = ws + 4 * NN;
  float* z   = ws + 5 * NN;

  __shared__ float sm_tiles[8 * 16 * TSTR];    // per-wave 16x64 weight tiles (34KB)
  __shared__ float sm_vn[NN];                  // normalized vector (stage A -> B,C)
  __shared__ float sm_a[NN];                   // k (stage C) / v2 (stage D)
  __shared__ float sm_b[NN];                   // u (stage C)
  __shared__ float smrA[8], smrB[8];
  __shared__ float sm_s0, sm_s1;               // mean/rstd, then kmax/kmin

  const int tid  = threadIdx.x;
  const int lane = tid & 31;
  const int wid  = tid >> 5;
  const int gw   = blockIdx.x * 8 + wid;       // global wave id, 0..1023
  float* mytile  = sm_tiles + wid * (16 * TSTR);

  for (int l = 0; l < NLAYER; ++l) {
    // ---- Stage A: block-local layernorm stats + normalized vn in LDS ----
    {
      float s = 0.f, s2 = 0.f;
#pragma unroll
      for (int p = 0; p < 4; ++p) {
        float xv = v[tid + 256 * p];
        s += xv; s2 += xv * xv;
      }
      s = wave_sum(s); s2 = wave_sum(s2);
      if (lane == 0) { smrA[wid] = s; smrB[wid] = s2; }
      __syncthreads();
      if (tid == 0) {
        float a = 0.f, b = 0.f;
#pragma unroll
        for (int i = 0; i < 8; ++i) { a += smrA[i]; b += smrB[i]; }
        float mean = a * (1.0f / NN);
        float var  = b * (1.0f / NN) - mean * mean;
        sm_s0 = mean;
        sm_s1 = rsqrtf(var + 1e-5f);
      }
      __syncthreads();
      float mean = sm_s0, rstd = sm_s1;
#pragma unroll
      for (int p = 0; p < 4; ++p) {
        int j = tid + 256 * p;
        float g  = nrm[(size_t)(l * 2 + 0) * NN + j];
        float bb = nrm[(size_t)(l * 2 + 1) * NN + j];
        sm_vn[j] = (v[j] - mean) * rstd * g + bb;
      }
      __syncthreads();
    }

    // ---- Stage B: q,k,u = Wq/Wk/Wv @ vn + bias  (WMMA GEMV, atomic accumulate) ----
    {
      const float* A0 = attn + (size_t)(l * 3) * NN * RS;
      for (int t = gw; t < 3072; t += NBLK * 8) { // 3 mats x 64 rowgroups x 16 kchunks
        int mi = t >> 10;                         // 0..2
        int rg = (t >> 4) & 63;
        int c  = t & 15;
        gemv16(A0 + (size_t)mi * NN * RS, rg * 16, c * 64,
               mytile, sm_vn, q + (size_t)mi * NN, c == 0);
      }
    }
    grid_barrier(bar);

    // ---- Stage C: attention rows + skip; one wave per row ----
    {
#pragma unroll
      for (int p = 0; p < 4; ++p) {
        int j = tid + 256 * p;
        sm_a[j] = kk_[j];
        sm_b[j] = u[j];
      }
      __syncthreads();
      float mx = -3.402823466e38f, mn = 3.402823466e38f;
#pragma unroll
      for (int p = 0; p < 4; ++p) {
        float kv = sm_a[tid + 256 * p];
        mx = fmaxf(mx, kv); mn = fminf(mn, kv);
      }
      mx = wave_max(mx); mn = wave_min(mn);
      if (lane == 0) { smrA[wid] = mx; smrB[wid] = mn; }
      __syncthreads();
      if (tid == 0) {
        float a = smrA[0], b = smrB[0];
#pragma unroll
        for (int i = 1; i < 8; ++i) { a = fmaxf(a, smrA[i]); b = fminf(b, smrB[i]); }
        sm_s0 = a; sm_s1 = b;
      }
      __syncthreads();
      int   i0 = blockIdx.x * 8 + wid;            // this wave's row
      float sc = q[i0] * 0.03125f;                // q_i / sqrt(N)
      float mrow = (sc > 0.f) ? sc * sm_s0 : sc * sm_s1;  // exact row max
      float se = 0.f, su = 0.f;
      for (int j = lane; j < NN; j += 32) {
        float e = __expf(sc * sm_a[j] - mrow);
        se += e;
        su += e * sm_b[j];
      }
      se = wave_sum(se); su = wave_sum(su);
      if (lane == 0) {
        v2[i0] = su / se + sm_vn[i0];
        z[i0]  = 0.f;                             // clear accumulator for stage D
      }
    }
    grid_barrier(bar);

    // ---- Stage D: z = Wb @ v2 + bias  (WMMA GEMV) ----
    {
#pragma unroll
      for (int p = 0; p < 4; ++p) sm_a[tid + 256 * p] = v2[tid + 256 * p];
      __syncthreads();
      const float* WB = w_b + (size_t)l * NN * RS;
      for (int t = gw; t < 1024; t += NBLK * 8) { // 64 rowgroups x 16 kchunks
        int rg = (t >> 4) & 63;
        int c  = t & 15;
        gemv16(WB, rg * 16, c * 64, mytile, sm_a, z, c == 0);
      }
    }
    grid_barrier(bar);

    // ---- Stage E: activation -> v (or identity -> dout), zero q/k/u ----
    {
      if (tid < 8) {
        int i = blockIdx.x * 8 + tid;
        float zv = z[i];
        if (l < NLAYER - 1) {
          float a0 = ada[((size_t)l * NN + i) * 2 + 0];
          float b0 = ada[((size_t)l * NN + i) * 2 + 1];
          float t  = b0 * zv;
          v[i] = a0 * t / (1.f + __expf(-t));     // a * silu(b*z)
        } else {
          dout[i] = zv;                            // output layer: identity
        }
      }
      if (tid < 24) q[blockIdx.x * 24 + tid] = 0.f; // q,k,u contiguous: 3072 floats
    }
    grid_barrier(bar);
  }
}

extern "C" void kernel_launch(void* const* d_in, const int* in_sizes, int n_in,
                              void* d_out, int out_size, void* d_ws, size_t ws_size,
                              hipStream_t stream) {
  const float* x    = (const float*)d_in[0];   // [1024]
  const float* w_b  = (const float*)d_in[1];   // [64,1024,1025]
  const float* attn = (const float*)d_in[2];   // [64,3,1024,1025]
  const float* nrm  = (const float*)d_in[3];   // [64,2,1024]
  const float* ada  = (const float*)d_in[4];   // [64,1024,2]

  float* ws = (float*)d_ws;
  unsigned int* bar = (unsigned int*)((char*)d_ws + (size_t)6 * NN * sizeof(float));

  // zero accumulators (q,k,u,z), v, and the barrier counter; then v <- x.
  hipMemsetAsync(d_ws, 0, (size_t)6 * NN * sizeof(float) + 256, stream);
  hipMemcpyAsync(ws, x, NN * sizeof(float), hipMemcpyDeviceToDevice, stream);

  net_kernel<<<NBLK, BLKT, 0, stream>>>(w_b, attn, nrm, ada,
                                        (float*)d_out, ws, bar);
}